// SmoothL1LossWordAndSentenceAndImage_77464030151305
// MI455X (gfx1250) — compile-verified
//
#include <hip/hip_runtime.h>
#include <hip/hip_bf16.h>

// SmoothL1 word+sentence+image loss, B=32, T=512, D=1024, f32.
// Pure bandwidth-bound streaming reduction: 128 MB @ 23.3 TB/s ~= 5.5 us floor.
// pass1: stream preds/word_targets once (NT b128 loads), per-(chunk,b) partials.
// pass2: fold chunks -> per-b sentence & image terms (incl. L2-normalize image).
// pass3: single wave sums all 320 pre-scaled partials via chained
//        V_WMMA_F32_16X16X4_F32 (B = ones) -> final scalar loss.

typedef float f32x4 __attribute__((ext_vector_type(4)));
typedef float f32x2 __attribute__((ext_vector_type(2)));
typedef float v8f   __attribute__((ext_vector_type(8)));

#define B_  32
#define T_  512
#define D_  1024
#define MAXCH 8
#define PART_FLOATS 320   // 256 word partials + 32 sentence + 32 image
#define WOFF_WORD 0
#define WOFF_SENT 256
#define WOFF_IMG  288
#define EPS_ 1e-12f

__device__ __forceinline__ float smoothl1(float x) {
    float a = fabsf(x);
    return (a < 1.0f) ? 0.5f * x * x : (a - 0.5f);
}

__device__ __forceinline__ float smoothl1_4(f32x4 v) {
    return smoothl1(v.x) + smoothl1(v.y) + smoothl1(v.z) + smoothl1(v.w);
}

// wave32 sum (CDNA5 is wave32-only)
__device__ __forceinline__ float waveSum(float v) {
#pragma unroll
    for (int off = 16; off > 0; off >>= 1)
        v += __shfl_xor(v, off, 32);
    return v;
}

// 256-thread (8-wave) block sum, result broadcast to all threads.
__device__ __forceinline__ float blockSumB(float v) {
    __shared__ float red[8];
    __shared__ float total;
    int lane = threadIdx.x & 31;
    int wid  = threadIdx.x >> 5;
    v = waveSum(v);
    __syncthreads();              // guard shared reuse across successive calls
    if (lane == 0) red[wid] = v;
    __syncthreads();
    if (wid == 0) {
        float r = (lane < 8) ? red[lane] : 0.0f;
        r = waveSum(r);
        if (lane == 0) total = r;
    }
    __syncthreads();
    return total;
}

__global__ void loss_init_kernel(float* __restrict__ ws) {
    if (threadIdx.x < PART_FLOATS) ws[threadIdx.x] = 0.0f;
}

// grid(CH, B), block(256). Each thread owns 4 consecutive d columns (b128 loads).
__global__ void loss_pass1_kernel(const float* __restrict__ word_targets,
                                  const float* __restrict__ preds,
                                  const int*   __restrict__ lens,
                                  float*       __restrict__ ws,
                                  int CH, int TCHUNK) {
    const int ch  = blockIdx.x;
    const int b   = blockIdx.y;
    const int tid = threadIdx.x;
    const int len = lens[b];
    const int t0  = ch * TCHUNK;
    const int t1  = min(t0 + TCHUNK, len);

    const f32x4* pB = (const f32x4*)(preds        + (size_t)b * T_ * D_);
    const f32x4* wB = (const f32x4*)(word_targets + (size_t)b * T_ * D_);

    f32x4 ps = {0.f, 0.f, 0.f, 0.f};
    f32x4 ts = {0.f, 0.f, 0.f, 0.f};
    float sl = 0.0f;
    for (int t = t0; t < t1; ++t) {
        // single-use streaming data: non-temporal 128-bit loads
        f32x4 pv = __builtin_nontemporal_load(&pB[(size_t)t * (D_ / 4) + tid]);
        f32x4 wv = __builtin_nontemporal_load(&wB[(size_t)t * (D_ / 4) + tid]);
        ps += pv;
        ts += wv;
        sl += smoothl1_4(pv - wv);
    }

    f32x4* predPart = (f32x4*)(ws + PART_FLOATS);
    f32x4* targPart = predPart + (size_t)CH * B_ * (D_ / 4);
    const size_t slot = ((size_t)ch * B_ + b) * (D_ / 4) + tid;
    predPart[slot] = ps;   // always written (zeros if chunk past len) -> deterministic
    targPart[slot] = ts;

    float tot = blockSumB(sl);
    if (tid == 0)
        ws[WOFF_WORD + b * CH + ch] =
            tot / ((float)len * (float)D_ * (float)B_);   // pre-scaled word partial
}

// grid(B), block(256): fold chunks, sentence + image terms per batch row.
__global__ void loss_pass2_kernel(const float* __restrict__ image_targets,
                                  const int*   __restrict__ lens,
                                  float*       __restrict__ ws,
                                  int CH) {
    const int b   = blockIdx.x;
    const int tid = threadIdx.x;
    const int len = lens[b];

    const f32x4* predPart = (const f32x4*)(ws + PART_FLOATS);
    const f32x4* targPart = predPart + (size_t)CH * B_ * (D_ / 4);

    f32x4 ps = {0.f, 0.f, 0.f, 0.f};
    f32x4 ts = {0.f, 0.f, 0.f, 0.f};
    for (int ch = 0; ch < CH; ++ch) {
        const size_t slot = ((size_t)ch * B_ + b) * (D_ / 4) + tid;
        ps += predPart[slot];
        ts += targPart[slot];
    }
    const float inv = 1.0f / (float)len;
    f32x4 pm = ps * inv;   // per-sentence mean of preds
    f32x4 tm = ts * inv;   // per-sentence mean of word targets

    f32x4 iv = ((const f32x4*)(image_targets + (size_t)b * D_))[tid];
    float nrm2 = blockSumB(iv.x * iv.x + iv.y * iv.y + iv.z * iv.z + iv.w * iv.w);
    float innorm = 1.0f / fmaxf(sqrtf(nrm2), EPS_);

    float sent = smoothl1_4(pm - tm);
    float im   = smoothl1_4(pm - iv * innorm);
    sent = blockSumB(sent);
    im   = blockSumB(im);
    if (tid == 0) {
        const float s = 1.0f / ((float)D_ * (float)B_);
        ws[WOFF_SENT + b] = sent * s;   // pre-scaled sentence partial
        ws[WOFF_IMG  + b] = im * s;     // pre-scaled image partial
    }
}

// single wave32: sum 320 pre-scaled partials with chained V_WMMA_F32_16X16X4_F32.
// B = ones(4x16) => D[i,j] = rowsum(A_i) for every column j; grand total of A =
// (sum of 8 C VGPRs in lane L) + (same in lane L+16). EXEC all-1s (32 threads).
__global__ void loss_pass3_kernel(const float* __restrict__ ws,
                                  float* __restrict__ out) {
    const int lane = threadIdx.x;      // 0..31
    v8f c = {0.f, 0.f, 0.f, 0.f, 0.f, 0.f, 0.f, 0.f};
    f32x2 ones = {1.0f, 1.0f};
#pragma unroll
    for (int i = 0; i < 5; ++i) {      // 5 x 64 = 320 partials, f32-exact
        f32x2 a;
        a.x = ws[i * 64 + 2 * lane];
        a.y = ws[i * 64 + 2 * lane + 1];
        // (neg_a, A, neg_b, B, c_mod, C, reuse_a, reuse_b)
        c = __builtin_amdgcn_wmma_f32_16x16x4_f32(false, a, false, ones,
                                                  (short)0, c, false, false);
    }
    float s = c[0] + c[1] + c[2] + c[3] + c[4] + c[5] + c[6] + c[7];
    s += __shfl_xor(s, 16, 32);        // lane L + lane L^16 => grand total (all lanes)
    if (lane == 0) out[0] = s;
}

extern "C" void kernel_launch(void* const* d_in, const int* in_sizes, int n_in,
                              void* d_out, int out_size, void* d_ws, size_t ws_size,
                              hipStream_t stream) {
    const float* word_targets  = (const float*)d_in[0];
    const float* image_targets = (const float*)d_in[1];
    const float* preds         = (const float*)d_in[2];
    const int*   decode_lens   = (const int*)d_in[3];
    float* ws  = (float*)d_ws;
    float* out = (float*)d_out;

    // Shrink chunk count if workspace is tight (needs 320 + 2*CH*B*D floats).
    const size_t availF = ws_size / sizeof(float);
    int CH = MAXCH;
    while (CH > 1 && (size_t)PART_FLOATS + 2ull * CH * B_ * D_ > availF) --CH;
    const int TCHUNK = (T_ + CH - 1) / CH;

    loss_init_kernel<<<1, PART_FLOATS, 0, stream>>>(ws);
    loss_pass1_kernel<<<dim3(CH, B_), 256, 0, stream>>>(word_targets, preds,
                                                        decode_lens, ws, CH, TCHUNK);
    loss_pass2_kernel<<<B_, 256, 0, stream>>>(image_targets, decode_lens, ws, CH);
    loss_pass3_kernel<<<1, 32, 0, stream>>>(ws, out);
}